// QuantumCircuit_87771951661496
// MI455X (gfx1250) — compile-verified
//
#include <hip/hip_runtime.h>
#include <math.h>

// ---------------------------------------------------------------------------
// 12-qubit batched state-vector simulator for MI455X (gfx1250).
//  - state per batch element: 4096 complex64 = Re/Im f32 planes in LDS
//    (padded: physical index = i + (i>>4) + 4*(i>>8) to kill bank conflicts)
//  - per layer: three 16x16 complex matmul passes (one per 4-qubit group)
//    executed with V_WMMA_F32_16X16X4_F32 chains; two tiles are processed
//    concurrently (4 interleaved accumulator chains) to hide WMMA RAW latency
//  - the 12-CNOT ring = fixed index permutation, folded into the scatter
//    stores of the (out-of-place) third pass via an LDS f^-1 table that
//    stores *physical* target offsets
//  All LDS traffic goes through integer offsets into one extern __shared__
//  array so the compiler keeps address space 3 (ds_* ops, not flat_*).
// ---------------------------------------------------------------------------

typedef float v2f __attribute__((ext_vector_type(2)));
typedef float v8f __attribute__((ext_vector_type(8)));

#define QC_BATCH_PER_BLOCK 4
#define QC_THREADS 256
#define QC_DIM 4096

#define QC_PLANE 4416                 // padded plane stride (>= 4410), 69*64
#define QC_BUF (2 * QC_PLANE)         // Re+Im planes: 8832 floats
#define QC_ELEM (2 * QC_BUF)          // ping+pong:   17664 floats
#define QC_FINV_OFF (QC_BATCH_PER_BLOCK * QC_ELEM)  // 70656
#define QC_LDS_FLOATS (QC_FINV_OFF + QC_DIM)        // 74752
#define QC_LDS_BYTES (QC_LDS_FLOATS * 4)            // 299008 (< 320K/WGP)

static __device__ __forceinline__ v8f wmma4(v2f a, v2f b, v8f c) {
  // D(16x16) += A(16x4) * B(4x16), f32
  return __builtin_amdgcn_wmma_f32_16x16x4_f32(false, a, false, b, (short)0, c,
                                               false, false);
}

// padded physical index for a logical state index 0..4095
static __device__ __forceinline__ int physidx(int i) {
  return i + (i >> 4) + 4 * (i >> 8);
}

// composite read-index function of the 12 sequential CNOTs (q -> (q+1)%12)
static __device__ __forceinline__ int fperm(int i) {
  int x = i;
#pragma unroll
  for (int q = 11; q >= 0; --q) {
    int C = 11 - q;
    int T = 11 - ((q + 1) % 12);
    x ^= ((x >> C) & 1) << T;
  }
  return x;
}

// One 16x16 complex matmul pass over the full state (one wave does 8 tiles,
// two tiles in flight per iteration).
// Logical element index:  idx = CTl*t + CMl*m + CKl*k   (disjoint bit fields)
// Physical (padded):      idx = CT*t  + CM*m  + CK*k
template <int PASS, bool PERM>
static __device__ __forceinline__ void apply_pass(float* lds, int srcOff,
                                                  int dstOff,
                                                  const float* wsMat, int lane,
                                                  int h) {
  constexpr int CT = (PASS == 0) ? 17 : 276;
  constexpr int CM = (PASS == 2) ? 17 : 1;
  constexpr int CK = (PASS == 0) ? 276 : ((PASS == 1) ? 17 : 1);

  // Pre-swizzled B operands for U^T: re, im, -im planes (global, L2-hot).
  float ur[8], ui[8], nui[8];
  const float* ubase = wsMat + lane * 8;
#pragma unroll
  for (int s = 0; s < 8; ++s) {
    ur[s] = ubase[s];
    ui[s] = ubase[256 + s];
    nui[s] = ubase[512 + s];
  }
  const int m = lane & 15;          // A row / D column (same lane field)
  const int kh = (lane >> 4) << 1;  // K half select for A/B layout
  const int rh = (lane >> 4) << 3;  // C/D row-half offset

  const int baseSrc = srcOff + CT * (h * 8) + CM * m + CK * kh;
  const int baseDst = dstOff + CT * (h * 8) + CM * rh + CK * m;
  const int baseDstLog = 256 * (h * 8) + 16 * rh + m;  // logical key (pass 2)

#pragma unroll 1
  for (int tp = 0; tp < 4; ++tp) {  // two tiles per iteration
    const int t0 = 2 * tp, t1 = 2 * tp + 1;
    float ar0[8], ai0[8], ar1[8], ai1[8];
#pragma unroll
    for (int j = 0; j < 4; ++j) {
      const int o0 = baseSrc + CT * t0 + CK * (4 * j);
      const int o1 = baseSrc + CT * t1 + CK * (4 * j);
      ar0[2 * j] = lds[o0];
      ar0[2 * j + 1] = lds[o0 + CK];
      ai0[2 * j] = lds[o0 + QC_PLANE];
      ai0[2 * j + 1] = lds[o0 + CK + QC_PLANE];
      ar1[2 * j] = lds[o1];
      ar1[2 * j + 1] = lds[o1 + CK];
      ai1[2 * j] = lds[o1 + QC_PLANE];
      ai1[2 * j + 1] = lds[o1 + CK + QC_PLANE];
    }
    v8f dr0 = {0.f, 0.f, 0.f, 0.f, 0.f, 0.f, 0.f, 0.f};
    v8f di0 = {0.f, 0.f, 0.f, 0.f, 0.f, 0.f, 0.f, 0.f};
    v8f dr1 = {0.f, 0.f, 0.f, 0.f, 0.f, 0.f, 0.f, 0.f};
    v8f di1 = {0.f, 0.f, 0.f, 0.f, 0.f, 0.f, 0.f, 0.f};
#pragma unroll
    for (int j = 0; j < 4; ++j) {
      v2f Br = {ur[2 * j], ur[2 * j + 1]};
      v2f Bi = {ui[2 * j], ui[2 * j + 1]};
      v2f Bn = {nui[2 * j], nui[2 * j + 1]};
      v2f Ar0 = {ar0[2 * j], ar0[2 * j + 1]};
      v2f Ai0 = {ai0[2 * j], ai0[2 * j + 1]};
      v2f Ar1 = {ar1[2 * j], ar1[2 * j + 1]};
      v2f Ai1 = {ai1[2 * j], ai1[2 * j + 1]};
      // S' = S * U^T (complex): Dr = Sr*UrT - Si*UiT ; Di = Sr*UiT + Si*UrT
      // 4 independent chains -> same-accumulator RAW distance = 4
      dr0 = wmma4(Ar0, Br, dr0);
      di0 = wmma4(Ar0, Bi, di0);
      dr1 = wmma4(Ar1, Br, dr1);
      di1 = wmma4(Ar1, Bi, di1);
      dr0 = wmma4(Ai0, Bn, dr0);
      di0 = wmma4(Ai0, Br, di0);
      dr1 = wmma4(Ai1, Bn, dr1);
      di1 = wmma4(Ai1, Br, di1);
    }
#pragma unroll
    for (int r = 0; r < 8; ++r) {
      if (PERM) {
        const int key0 = baseDstLog + 256 * t0 + 16 * r;
        const int key1 = baseDstLog + 256 * t1 + 16 * r;
        const int ph0 = __float_as_int(lds[QC_FINV_OFF + key0]);
        const int ph1 = __float_as_int(lds[QC_FINV_OFF + key1]);
        lds[dstOff + ph0] = dr0[r];
        lds[dstOff + ph0 + QC_PLANE] = di0[r];
        lds[dstOff + ph1] = dr1[r];
        lds[dstOff + ph1 + QC_PLANE] = di1[r];
      } else {
        const int d0 = baseDst + CT * t0 + CM * r;
        const int d1 = baseDst + CT * t1 + CM * r;
        lds[d0] = dr0[r];
        lds[d0 + QC_PLANE] = di0[r];
        lds[d1] = dr1[r];
        lds[d1 + QC_PLANE] = di1[r];
      }
    }
  }
}

// ---------------------------------------------------------------------------
// Setup kernel: build the 9 fused 16x16 complex group matrices (3 layers x
// 3 groups of 4 qubits), store U^T pre-swizzled into the WMMA B-operand lane
// layout: slot(j,p): element (k = 4j+p+2*(lane/16), n = lane%16) of U^T.
// ws layout per matrix: [re 256][im 256][-im 256] floats -> 768 floats.
// ---------------------------------------------------------------------------
__global__ void qc_setup_gates(const float* __restrict__ params,
                               float* __restrict__ ws) {
  const int tid = threadIdx.x;
  const int mat = tid >> 5;
  const int lane = tid & 31;
  if (mat >= 9) return;
  const int L = mat / 3, g = mat % 3;

  // fused single-qubit matrices M = RZ * RY * RX for the 4 qubits of group g
  float mr[4][2][2], mi[4][2][2];
#pragma unroll
  for (int s = 0; s < 4; ++s) {
    const int q = g * 4 + s;
    const int base = L * 36 + q * 3;
    const float tx = params[base], ty = params[base + 1], tz = params[base + 2];
    const float cx = cosf(0.5f * tx), sx = sinf(0.5f * tx);
    const float cy = cosf(0.5f * ty), sy = sinf(0.5f * ty);
    const float cz = cosf(0.5f * tz), sz = sinf(0.5f * tz);
    // RY*RX = [[cycx + i sysx, -sycx - i cysx], [sycx - i cysx, cycx - i sysx]]
    float rr, ii;
    rr = cy * cx; ii = sy * sx;            // row0 * (cz - i sz)
    mr[s][0][0] = cz * rr + sz * ii; mi[s][0][0] = cz * ii - sz * rr;
    rr = -sy * cx; ii = -cy * sx;
    mr[s][0][1] = cz * rr + sz * ii; mi[s][0][1] = cz * ii - sz * rr;
    rr = sy * cx; ii = -cy * sx;           // row1 * (cz + i sz)
    mr[s][1][0] = cz * rr - sz * ii; mi[s][1][0] = cz * ii + sz * rr;
    rr = cy * cx; ii = -sy * sx;
    mr[s][1][1] = cz * rr - sz * ii; mi[s][1][1] = cz * ii + sz * rr;
  }

  float* base = ws + mat * 768;
  const int n = lane & 15;
  const int khalf = (lane >> 4) << 1;
#pragma unroll
  for (int j = 0; j < 4; ++j) {
#pragma unroll
    for (int p = 0; p < 2; ++p) {
      const int k = 4 * j + p + khalf;
      // value = U^T[k][n] = U[n][k] = prod_s M_s[bit(n,3-s)][bit(k,3-s)]
      float re = 1.f, im = 0.f;
#pragma unroll
      for (int s = 0; s < 4; ++s) {
        const int rb = (n >> (3 - s)) & 1;
        const int cb = (k >> (3 - s)) & 1;
        const float wr = mr[s][rb][cb], wi = mi[s][rb][cb];
        const float nre = re * wr - im * wi;
        const float nim = re * wi + im * wr;
        re = nre; im = nim;
      }
      const int slot = lane * 8 + j * 2 + p;
      base[slot] = re;
      base[256 + slot] = im;
      base[512 + slot] = -im;
    }
  }
}

// ---------------------------------------------------------------------------
// Main kernel: 256 threads = 8 wave32; 4 batch elements per block,
// 2 waves per element (each covers 8 of 16 outer tiles per pass).
// ---------------------------------------------------------------------------
__global__ __launch_bounds__(QC_THREADS) void qc_main(
    const float* __restrict__ x, const float* __restrict__ ws,
    float* __restrict__ out) {
  extern __shared__ float lds[];
  const int tid = threadIdx.x;
  const int lane = tid & 31;
  const int w = tid >> 5;        // wave 0..7
  const int e = w >> 1;          // batch sub-element 0..3
  const int h = w & 1;           // tile half
  const int lt = h * 32 + lane;  // thread 0..63 within element
  const long long b = (long long)blockIdx.x * QC_BATCH_PER_BLOCK + e;

  const int eb = e * QC_ELEM;  // element base offset (floats)

  // Build inverse-permutation table: finv[f(p)] = physical(p).
  for (int p = tid; p < QC_DIM; p += QC_THREADS)
    lds[QC_FINV_OFF + fperm(p)] = __int_as_float(physidx(p));

  // Load x row into registers + accumulate sum of squares.
  const float4* xr = (const float4*)(x + b * QC_DIM);
  float4 v[16];
  float ss = 0.f;
#pragma unroll
  for (int ii = 0; ii < 16; ++ii) {
    v[ii] = xr[ii * 64 + lt];
    ss += v[ii].x * v[ii].x + v[ii].y * v[ii].y + v[ii].z * v[ii].z +
          v[ii].w * v[ii].w;
  }
#pragma unroll
  for (int off = 16; off > 0; off >>= 1) ss += __shfl_xor(ss, off);
  if (lane == 0) lds[eb + QC_PLANE + h] = ss;  // stash in (soon-zeroed) Im
  __syncthreads();
  const float inv = rsqrtf(lds[eb + QC_PLANE] + lds[eb + QC_PLANE + 1]);
  __syncthreads();
  // Scatter normalized amplitudes into padded Re plane; zero Im plane.
#pragma unroll
  for (int ii = 0; ii < 16; ++ii) {
    const int u = ii * 64 + lt;                  // logical float4 index
    const int pb = 4 * u + (u >> 2) + 4 * ii;    // physidx(4u); +j stays exact
    lds[eb + pb + 0] = v[ii].x * inv;
    lds[eb + pb + 1] = v[ii].y * inv;
    lds[eb + pb + 2] = v[ii].z * inv;
    lds[eb + pb + 3] = v[ii].w * inv;
    lds[eb + QC_PLANE + pb + 0] = 0.f;
    lds[eb + QC_PLANE + pb + 1] = 0.f;
    lds[eb + QC_PLANE + pb + 2] = 0.f;
    lds[eb + QC_PLANE + pb + 3] = 0.f;
  }
  __syncthreads();

  int cb = 0;
#pragma unroll 1
  for (int L = 0; L < 3; ++L) {
    const float* wsL = ws + L * 3 * 768;
    const int cur = eb + cb * QC_BUF;
    const int nxt = eb + (cb ^ 1) * QC_BUF;
    // group A (qubits 0-3), in place
    apply_pass<0, false>(lds, cur, cur, wsL + 0, lane, h);
    __syncthreads();
    // group B (qubits 4-7), in place
    apply_pass<1, false>(lds, cur, cur, wsL + 768, lane, h);
    __syncthreads();
    // group C (qubits 8-11) + CNOT-ring permutation, out of place
    apply_pass<2, true>(lds, cur, nxt, wsL + 1536, lane, h);
    __syncthreads();
    cb ^= 1;
  }
  const int fin = eb + cb * QC_BUF;

  // Expectation values <Z_q> = sum_i |amp_i|^2 * (1 - 2*bit_{11-q}(i)).
  // i = lt + 64*ii: bits 5:0 from lt (sign constant per thread, q>=6),
  //                 bits 11:6 from ii (sign constant per unrolled iter, q<6).
  const int ebase = fin + lt + (lt >> 4);
  float T = 0.f;
  float accL[6];
#pragma unroll
  for (int q = 0; q < 6; ++q) accL[q] = 0.f;
#pragma unroll
  for (int ii = 0; ii < 64; ++ii) {
    const int d = 68 * ii + 4 * (ii >> 2);  // physidx delta for i = lt + 64*ii
    const float re = lds[ebase + d];
    const float im = lds[ebase + d + QC_PLANE];
    const float p = re * re + im * im;
    T += p;
#pragma unroll
    for (int q = 0; q < 6; ++q)
      accL[q] += ((ii >> (5 - q)) & 1) ? -p : p;
  }
  float acc[12];
#pragma unroll
  for (int q = 0; q < 6; ++q) acc[q] = accL[q];
#pragma unroll
  for (int q = 6; q < 12; ++q)
    acc[q] = ((lt >> (11 - q)) & 1) ? -T : T;
#pragma unroll
  for (int q = 0; q < 12; ++q) {
    float s = acc[q];
#pragma unroll
    for (int off = 16; off > 0; off >>= 1) s += __shfl_xor(s, off);
    acc[q] = s;
  }
  __syncthreads();  // finv region dead -> reuse as reduction scratch
  if (lane == 0) {
#pragma unroll
    for (int q = 0; q < 12; ++q) lds[QC_FINV_OFF + w * 12 + q] = acc[q];
  }
  __syncthreads();
  if (h == 0 && lane < 12)
    out[b * 12 + lane] = lds[QC_FINV_OFF + (2 * e) * 12 + lane] +
                         lds[QC_FINV_OFF + (2 * e + 1) * 12 + lane];
}

// ---------------------------------------------------------------------------
extern "C" void kernel_launch(void* const* d_in, const int* in_sizes, int n_in,
                              void* d_out, int out_size, void* d_ws,
                              size_t ws_size, hipStream_t stream) {
  (void)n_in; (void)out_size; (void)ws_size;
  const float* x = (const float*)d_in[0];
  const float* params = (const float*)d_in[1];
  float* out = (float*)d_out;
  float* ws = (float*)d_ws;  // 9 * 768 floats = 27648 B of gate matrices

  const int batch = in_sizes[0] / QC_DIM;
  const int blocks = batch / QC_BATCH_PER_BLOCK;

  qc_setup_gates<<<1, 288, 0, stream>>>(params, ws);

  (void)hipFuncSetAttribute((const void*)qc_main,
                            hipFuncAttributeMaxDynamicSharedMemorySize,
                            QC_LDS_BYTES);
  qc_main<<<blocks, QC_THREADS, QC_LDS_BYTES, stream>>>(x, ws, out);
}